// Matcher_7026566496623
// MI455X (gfx1250) — compile-verified
//
#include <hip/hip_runtime.h>
#include <stdint.h>

#define B_  8
#define HW_ 3136                 // 56*56
#define ROWPITCH_B 12544         // HW_*4 bytes, multiple of 16
#define MATELEMS ((size_t)HW_*(size_t)HW_)
#define NJT 49                   // HW_/64 column tiles
#define NCHUNK 196               // HW_/16 row iterations (16 rows per block iter, 2 per wave)

// merge two descending sorted 4-lists -> top-4 of union (bitonic merge, 12 min/max)
__device__ __forceinline__ void merge4(float&a0,float&a1,float&a2,float&a3,
                                       float b0,float b1,float b2,float b3){
  float c0=fmaxf(a0,b3), c1=fmaxf(a1,b2), c2=fmaxf(a2,b1), c3=fmaxf(a3,b0);
  float d0=fmaxf(c0,c2), d2=fminf(c0,c2);
  float d1=fmaxf(c1,c3), d3=fminf(c1,c3);
  a0=fmaxf(d0,d1); a1=fminf(d0,d1);
  a2=fmaxf(d2,d3); a3=fminf(d2,d3);
}

// insert v into descending top-4 (guarded compare-exchange cascade)
__device__ __forceinline__ void ins4(float&t0,float&t1,float&t2,float&t3,float v){
  if (v > t3) {
    float s=v, n;
    n=fmaxf(t0,s); s=fminf(t0,s); t0=n;
    n=fmaxf(t1,s); s=fminf(t1,s); t1=n;
    n=fmaxf(t2,s); s=fminf(t2,s); t2=n;
    t3=fmaxf(t3,s);
  }
}

// -------- Phase A: per-row 4th-largest (cut) and min of raw prev_sim --------
__global__ __launch_bounds__(256) void row_stats_kernel(const float* __restrict__ mat,
                                                        float* __restrict__ cutw,
                                                        float* __restrict__ mnw){
  const int row = blockIdx.x;                       // b*HW_ + i
  const float4* rp = (const float4*)(mat + (size_t)row * HW_);
  float t0=-INFINITY,t1=-INFINITY,t2=-INFINITY,t3=-INFINITY, mn=INFINITY;
  for (int q = threadIdx.x; q < HW_/4; q += 256) {
    float4 v = rp[q];
    mn = fminf(mn, fminf(fminf(v.x,v.y), fminf(v.z,v.w)));
    ins4(t0,t1,t2,t3,v.x); ins4(t0,t1,t2,t3,v.y);
    ins4(t0,t1,t2,t3,v.z); ins4(t0,t1,t2,t3,v.w);
  }
  // wave32 xor-reduction: every lane ends with the wave's top-4 / min
  #pragma unroll
  for (int m=1; m<32; m<<=1) {
    float b0=__shfl_xor(t0,m), b1=__shfl_xor(t1,m);
    float b2=__shfl_xor(t2,m), b3=__shfl_xor(t3,m);
    mn = fminf(mn, __shfl_xor(mn,m));
    merge4(t0,t1,t2,t3,b0,b1,b2,b3);
  }
  __shared__ float sm[8][5];
  const int wid = threadIdx.x >> 5, lane = threadIdx.x & 31;
  if (lane==0){ sm[wid][0]=t0; sm[wid][1]=t1; sm[wid][2]=t2; sm[wid][3]=t3; sm[wid][4]=mn; }
  __syncthreads();
  if (threadIdx.x==0){
    float a0=sm[0][0],a1=sm[0][1],a2=sm[0][2],a3=sm[0][3], m2=sm[0][4];
    #pragma unroll
    for (int w=1; w<8; ++w){
      merge4(a0,a1,a2,a3, sm[w][0],sm[w][1],sm[w][2],sm[w][3]);
      m2=fminf(m2, sm[w][4]);
    }
    cutw[row]=a3;    // 4th largest of raw row
    mnw[row]=m2;     // min of raw row
  }
}

// -------- Phase B: streaming column-max via async global->LDS pipeline --------
// block = 256 threads = 8 waves; each wave streams 2 rows/chunk, triple-buffered.
template<bool LOCAL>
__global__ __launch_bounds__(256) void colmax_kernel(const float* __restrict__ mat,
                                                     const float* __restrict__ seg,   // (B,2,HW)
                                                     const float* __restrict__ cutw,
                                                     const float* __restrict__ mnw,
                                                     float* __restrict__ out,         // (B,4,HW)
                                                     int outch){
  __shared__ __align__(16) float abuf[8*3*128];     // 8 waves x 3 bufs x 512B = 12 KB
  __shared__ float p_s0[HW_], p_s1[HW_];
  __shared__ float p_ct[LOCAL ? HW_ : 1], p_mn[LOCAL ? HW_ : 1];

  const int b  = blockIdx.x / NJT;
  const int jt = blockIdx.x % NJT;
  const int j0 = jt * 64;

  for (int q = threadIdx.x; q < HW_; q += 256) {
    p_s0[q] = seg[(size_t)b*2*HW_ + q];
    p_s1[q] = seg[(size_t)b*2*HW_ + HW_ + q];
    if (LOCAL) { p_ct[q] = cutw[(size_t)b*HW_ + q]; p_mn[q] = mnw[(size_t)b*HW_ + q]; }
  }
  __syncthreads();

  const int wid  = threadIdx.x >> 5;
  const int lane = threadIdx.x & 31;
  const int rck  = lane >> 4;          // row-in-chunk this lane loads (0/1)
  const int c16  = lane & 15;          // 16B unit within the 256B row slice

  const uint64_t sbase = (uint64_t)(uintptr_t)(mat + (size_t)b * MATELEMS);
  const uint32_t ldsl  = (uint32_t)(uintptr_t)(&abuf[wid*384])
                       + (uint32_t)(rck*256 + c16*16);
  const uint32_t vo0   = (uint32_t)((wid*2 + rck)*ROWPITCH_B + j0*4 + c16*16);

  // prologue: issue chunks 0 and 1
  {
    uint32_t l0 = ldsl;
    asm volatile("global_load_async_to_lds_b128 %0, %1, %2"
                 :: "v"(l0), "v"(vo0), "s"(sbase) : "memory");
    uint32_t l1 = ldsl + 512u;
    uint32_t v1 = vo0 + 16u*ROWPITCH_B;
    asm volatile("global_load_async_to_lds_b128 %0, %1, %2"
                 :: "v"(l1), "v"(v1), "s"(sbase) : "memory");
  }

  float a00=-INFINITY, a01=-INFINITY, a10=-INFINITY, a11=-INFINITY;

  for (int k = 0; k < NCHUNK; ++k) {
    if (k + 2 < NCHUNK) {
      uint32_t li = ldsl + (uint32_t)(((k+2)%3)*512);
      uint32_t vi = vo0 + (uint32_t)(k+2)*(16u*ROWPITCH_B);
      asm volatile("global_load_async_to_lds_b128 %0, %1, %2"
                   :: "v"(li), "v"(vi), "s"(sbase) : "memory");
      asm volatile("s_wait_asynccnt 2" ::: "memory");   // oldest (chunk k) complete
    } else {
      asm volatile("s_wait_asynccnt 0" ::: "memory");
    }
    const float* cb = &abuf[wid*384 + (k%3)*128];
    const int i0 = k*16 + wid*2;
    #pragma unroll
    for (int r = 0; r < 2; ++r) {
      const int i = i0 + r;
      const float s0 = p_s0[i], s1 = p_s1[i];
      float c0v=0.f, c1v=0.f, m0v=0.f, m1v=0.f;
      if (LOCAL) {
        const float ct = p_ct[i], mv = p_mn[i];
        c0v = s0*ct; m0v = s0*mv;         // scaled cut / min: bit-exact vs reference
        c1v = s1*ct; m1v = s1*mv;
      }
      const float r0 = cb[r*64 + lane];         // conflict-free: addr%64 == lane
      const float r1 = cb[r*64 + lane + 32];
      float v00=r0*s0, v10=r0*s1, v01=r1*s0, v11=r1*s1;
      if (LOCAL) {
        v00 = (v00 < c0v) ? m0v : v00;
        v01 = (v01 < c0v) ? m0v : v01;
        v10 = (v10 < c1v) ? m1v : v10;
        v11 = (v11 < c1v) ? m1v : v11;
      }
      a00=fmaxf(a00,v00); a01=fmaxf(a01,v01);
      a10=fmaxf(a10,v10); a11=fmaxf(a11,v11);
    }
  }

  // cross-wave max, reusing the (now idle) async buffers
  {
    float* rw = &abuf[wid*384];
    rw[lane]       = a00;  rw[lane+32]  = a01;   // ch0 cols [lane], [lane+32]
    rw[64 + lane]  = a10;  rw[96 + lane] = a11;  // ch1
  }
  __syncthreads();
  if (threadIdx.x < 128) {
    const int ch  = threadIdx.x >> 6;
    const int col = threadIdx.x & 63;
    float m = abuf[ch*64 + col];
    #pragma unroll
    for (int w=1; w<8; ++w) m = fmaxf(m, abuf[w*384 + ch*64 + col]);
    out[((size_t)b*4 + outch + ch)*HW_ + j0 + col] = m;
  }
}

extern "C" void kernel_launch(void* const* d_in, const int* in_sizes, int n_in,
                              void* d_out, int out_size, void* d_ws, size_t ws_size,
                              hipStream_t stream) {
  const float* init_sim = (const float*)d_in[0];
  const float* prev_sim = (const float*)d_in[1];
  const float* init_seg = (const float*)d_in[2];
  const float* prev_seg = (const float*)d_in[3];
  float* out  = (float*)d_out;
  float* cutw = (float*)d_ws;                    // B*HW floats
  float* mnw  = cutw + (size_t)B_*HW_;           // B*HW floats (200 KB total)

  row_stats_kernel<<<B_*HW_, 256, 0, stream>>>(prev_sim, cutw, mnw);
  colmax_kernel<false><<<B_*NJT, 256, 0, stream>>>(init_sim, init_seg,
                                                   nullptr, nullptr, out, 0);
  colmax_kernel<true ><<<B_*NJT, 256, 0, stream>>>(prev_sim, prev_seg,
                                                   cutw, mnw, out, 2);
}